// AttentionAggregation_40046275067969
// MI455X (gfx1250) — compile-verified
//
#include <hip/hip_runtime.h>
#include <hip/hip_bf16.h>

// ---------------------------------------------------------------------------
// AttentionAggregation for MI455X (gfx1250, wave32)
//   v   = x @ W                      (f32 WMMA 16x16x4, full precision)
//   out = segment_sum(alpha * v[idx_j], idx_i)
// v (25.6MB) is L2-resident -> gather/scatter phase is L2 bandwidth + f32
// atomic bound; GEMM (1.6 GFLOP) is a trivial prologue on the WMMA pipe.
// Scatter atomics are emitted directly as global_atomic_add_f32 via inline
// asm (non-returning HW f32 atomic, STOREcnt-tracked, resolved in L2).
// ---------------------------------------------------------------------------

typedef __attribute__((ext_vector_type(2))) float v2f;
typedef __attribute__((ext_vector_type(4))) float v4f;
typedef __attribute__((ext_vector_type(8))) float v8f;

#define FEAT 128

// Guaranteed CDNA5 hardware f32 atomic add (no CAS loop possible).
__device__ __forceinline__ void hw_atomic_add_f32(float* p, float v) {
  asm volatile("global_atomic_add_f32 %0, %1, off" : : "v"(p), "v"(v) : "memory");
}

// ---------------------------------------------------------------------------
// Kernel 3 (bottleneck): out[idx_i[p]] += alpha[p] * v[idx_j[p]], one wave/pair.
// Each lane: one global_load_b128 of v (L2 hit) + 4 global_atomic_add_f32.
// ---------------------------------------------------------------------------
__global__ __launch_bounds__(256) void gather_scale_scatter(
    const float* __restrict__ v, const float* __restrict__ alpha,
    const int* __restrict__ idx_i, const int* __restrict__ idx_j,
    float* __restrict__ out, int n_pairs) {
  const int wave = threadIdx.x >> 5;
  const int lane = threadIdx.x & 31;
  const int p = blockIdx.x * 8 + wave;
  if (p >= n_pairs) return;

  const float a = alpha[p];
  const int j = idx_j[p];
  const int i = idx_i[p];

  const v4f val = *(const v4f*)(v + (size_t)j * FEAT + lane * 4);  // L2 hit
  float* dst = out + (size_t)i * FEAT + lane * 4;

  hw_atomic_add_f32(dst + 0, a * val.x);
  hw_atomic_add_f32(dst + 1, a * val.y);
  hw_atomic_add_f32(dst + 2, a * val.z);
  hw_atomic_add_f32(dst + 3, a * val.w);
}

// ---------------------------------------------------------------------------
// Kernel 1: v = x @ W with V_WMMA_F32_16X16X4_F32.
// One wave -> one 16x128 strip of v (8 v8f accumulators). A fragment loaded
// ONCE per k-step (global_load_b64) and reused by 8 WMMAs -> x read once.
// W staged in LDS in PAIRED-K layout:
//     Wlds[(k>>1)*256 + n*2 + (k&1)]
// so a B fragment (W[k][n], W[k+1][n]) is one aligned ds_load_b64 into a
// consecutive VGPR pair -- removes the v_dual_mov shuffles seen in round 2.
// A-matrix 16x4 f32 layout (ISA 7.12.2): lanes 0-15: (M=lane, K=0,1 in v0,v1)
//                                        lanes 16-31: (M=lane-16, K=2,3)
// B-matrix 4x16: v0 = (K=0|2, N=lane&15), v1 = (K=1|3, N=lane&15)
// C/D 16x16: VGPR r: lanes 0-15 -> (M=r, N=lane); lanes 16-31 -> (M=r+8, N-16)
// ---------------------------------------------------------------------------
__global__ __launch_bounds__(256) void gemm_xw_wmma(
    const float* __restrict__ x, const float* __restrict__ W,
    float* __restrict__ v, int n_nodes) {
  __shared__ float Wlds[FEAT * FEAT];  // 64 KB of the WGP's 320 KB

  // stage W into paired-K layout (one-time, coalesced global reads)
  for (int i = threadIdx.x; i < FEAT * FEAT; i += 256) {
    const int k = i >> 7;        // i / 128
    const int n = i & 127;       // i % 128
    Wlds[(k >> 1) * 256 + n * 2 + (k & 1)] = W[i];
  }
  __syncthreads();

  const int wave  = threadIdx.x >> 5;
  const int mTile = blockIdx.x * 8 + wave;     // one 16-row strip per wave
  const int mBase = mTile * 16;
  if (mBase >= n_nodes) return;                // wave-uniform: EXEC all-1s

  const int lane = threadIdx.x & 31;
  const int half = lane >> 4;                  // 0: K pair {0,1}, 1: {2,3}
  const int l16  = lane & 15;
  const int koff = half * 2;
  const float* __restrict__ xrow = x + (size_t)(mBase + l16) * FEAT;

  v8f acc[8];
  #pragma unroll
  for (int n = 0; n < 8; ++n) acc[n] = (v8f){};

  #pragma unroll 2
  for (int k0 = 0; k0 < FEAT; k0 += 4) {
    // A fragment: one contiguous float2 per lane (global_load_b64), reused 8x
    const v2f a = *(const v2f*)(xrow + k0 + koff);
    // paired-K base for this lane's K pair: ((k0+koff)/2) * 256
    const float* __restrict__ wrow = Wlds + ((k0 + koff) >> 1) * 256;
    #pragma unroll
    for (int n = 0; n < 8; ++n) {
      // single aligned ds_load_b64 -> consecutive VGPR pair
      const v2f b = *(const v2f*)(wrow + (n * 16 + l16) * 2);
      acc[n] = __builtin_amdgcn_wmma_f32_16x16x4_f32(
          false, a, false, b, (short)0, acc[n], false, false);
    }
  }

  // store the 16x128 strip; b32 stores coalesce across lanes 0-15 / 16-31
  #pragma unroll
  for (int n = 0; n < 8; ++n) {
    #pragma unroll
    for (int r = 0; r < 8; ++r) {
      const int m = mBase + half * 8 + r;
      v[(size_t)m * FEAT + n * 16 + l16] = acc[n][r];
    }
  }
}

// ---------------------------------------------------------------------------
// Kernel 2: zero the output accumulator (harness poisons d_out with 0xAA).
// ---------------------------------------------------------------------------
__global__ void zero_out(float* __restrict__ out, int n) {
  const int i = blockIdx.x * blockDim.x + threadIdx.x;
  if (i < n) out[i] = 0.0f;
}

// ---------------------------------------------------------------------------
// Launcher. Inputs: x, alpha_ij, idx_i, idx_j, W. Output: (n_nodes, 128) f32.
// d_ws holds v = x @ W (n_nodes * 128 * 4 B = 25.6 MB).
// ---------------------------------------------------------------------------
extern "C" void kernel_launch(void* const* d_in, const int* in_sizes, int n_in,
                              void* d_out, int out_size, void* d_ws, size_t ws_size,
                              hipStream_t stream) {
  const float* x     = (const float*)d_in[0];
  const float* alpha = (const float*)d_in[1];
  const int*   idx_i = (const int*)d_in[2];
  const int*   idx_j = (const int*)d_in[3];
  const float* W     = (const float*)d_in[4];
  float* out = (float*)d_out;
  float* v   = (float*)d_ws;

  const int n_nodes = in_sizes[0] / FEAT;   // 50000
  const int n_pairs = in_sizes[1];          // 625000

  // 1) v = x @ W : one 16x128 strip per wave, 8 waves per block
  const int m_tiles = (n_nodes + 15) / 16;  // 3125 (exact)
  gemm_xw_wmma<<<(m_tiles + 7) / 8, 256, 0, stream>>>(x, W, v, n_nodes);

  // 2) out = 0
  const int n_out = n_nodes * FEAT;
  zero_out<<<(n_out + 255) / 256, 256, 0, stream>>>(out, n_out);

  // 3) scatter-add alpha-scaled gathered rows
  gather_scale_scatter<<<(n_pairs + 7) / 8, 256, 0, stream>>>(
      v, alpha, idx_i, idx_j, out, n_pairs);
}